// SE3CouplingLayer_49409303773590
// MI455X (gfx1250) — compile-verified
//
#include <hip/hip_runtime.h>
#include <hip/hip_bf16.h>
#include <math.h>

// ---------------------------------------------------------------------------
// SE3 coupling layer for MI455X (gfx1250, wave32).
// All GEMMs (neigh = w@feats, H x H scalar-net / head GEMMs) run on
// V_WMMA_F32_16X16X4_F32 with f32 accumulation (exact vs f32 reference).
// The (B,s,s) weight matrix w is never materialized: each block computes a
// 16x64 tile of w in LDS from coordinates and streams WMMA over K.
// ---------------------------------------------------------------------------

typedef float v2f __attribute__((ext_vector_type(2)));
typedef float v8f __attribute__((ext_vector_type(8)));

#define HD   128          // feature width
#define SVAL 1024         // atoms per half (s)
#define BVAL 16           // batch
#define BK   64           // K-chunk for neigh GEMM
#define MROWS (BVAL*SVAL) // 16384 rows for HxH GEMMs

__device__ __forceinline__ float geluf(float x) {
  // jax.nn.gelu default (approximate=True, tanh form)
  const float k0 = 0.7978845608028654f;
  float x3 = x * x * x;
  return 0.5f * x * (1.0f + tanhf(k0 * (x + 0.044715f * x3)));
}

__device__ __forceinline__ float softplusf(float x) {
  return (x > 20.0f) ? x : log1pf(__expf(x));
}

// ---------------------------------------------------------------------------
// feats[b,i,:] = embed[types[b,i],:]   (first half atoms only)
// ---------------------------------------------------------------------------
__global__ void k_embed(const int* __restrict__ types,
                        const float* __restrict__ embed,
                        float* __restrict__ feats, int total) {
  int idx = blockIdx.x * blockDim.x + threadIdx.x;
  if (idx >= total) return;
  int h = idx & (HD - 1);
  int a = idx >> 7;              // b*s + i
  int b = a / SVAL;
  int i = a - b * SVAL;
  int t = types[b * 2 * SVAL + i];
  feats[idx] = embed[t * HD + h];
}

// ---------------------------------------------------------------------------
// combined = (feats + w @ feats) * mask_row ; w computed on the fly.
// Block: 256 thr (8 waves). Block owns (batch b, 16-row tile). Wave wv owns
// output columns [16*wv, 16*wv+16). K loop in chunks of 64.
// ---------------------------------------------------------------------------
__global__ __launch_bounds__(256) void k_neigh(
    const float* __restrict__ coords,          // (B, 2s, 3)
    const unsigned char* __restrict__ mask,    // (B, 2s) bool
    const float* __restrict__ feats,           // (B*s, 128)
    float* __restrict__ combined) {            // (B*s, 128)
  __shared__ float rX[16], rY[16], rZ[16], rMf[16];
  __shared__ float wt[16][68];    // padded: bank = (4r + c) & 63
  __shared__ float ft[BK][132];   // padded: bank = (4k + c) & 63

  const int tid = threadIdx.x;
  const int ntile = SVAL / 16;              // 64 row-tiles per batch
  const int b  = blockIdx.x / ntile;
  const int m0 = (blockIdx.x - b * ntile) * 16;
  const int N2 = 2 * SVAL;

  if (tid < 16) {
    int gi = b * N2 + m0 + tid;
    rX[tid]  = coords[gi * 3 + 0];
    rY[tid]  = coords[gi * 3 + 1];
    rZ[tid]  = coords[gi * 3 + 2];
    rMf[tid] = mask[gi] ? 1.0f : 0.0f;
  }
  __syncthreads();

  const int lane = tid & 31;
  const int wv   = tid >> 5;
  const int half = lane >> 4;
  const int r    = lane & 15;
  const int n0   = wv * 16;

  // per-thread assignment for w-tile generation: row wi, 4 cols at wk0
  const int wi  = tid & 15;
  const int wk0 = (tid >> 4) * 4;
  const float rx = rX[wi], ry = rY[wi], rz = rZ[wi];
  const bool  rm = (rMf[wi] != 0.0f);

  v8f acc = {0.f,0.f,0.f,0.f,0.f,0.f,0.f,0.f};

  for (int k0 = 0; k0 < SVAL; k0 += BK) {
    // ---- build 16 x 64 w tile ----
    #pragma unroll
    for (int t = 0; t < 4; ++t) {
      int kk = wk0 + t;
      int gj = b * N2 + k0 + kk;
      float dx = rx - coords[gj * 3 + 0];
      float dy = ry - coords[gj * 3 + 1];
      float dz = rz - coords[gj * 3 + 2];
      float d  = sqrtf(dx * dx + dy * dy + dz * dz + 1e-12f);
      bool valid = rm && (mask[gj] != 0);
      d = valid ? d : 1000000.0f;
      wt[wi][kk] = (d < 5.0f) ? __expf(-d * 0.2f) : 0.0f;
    }
    // ---- stage 64 x 128 feats tile (2048 float4 / 256 thr = 8 each) ----
    const float4* f4 = (const float4*)(feats + (size_t)(b * SVAL + k0) * HD);
    #pragma unroll
    for (int t = 0; t < 8; ++t) {
      int idx = tid + t * 256;
      int fr = idx >> 5, fc = idx & 31;
      float4 v = f4[fr * 32 + fc];
      ft[fr][fc * 4 + 0] = v.x; ft[fr][fc * 4 + 1] = v.y;
      ft[fr][fc * 4 + 2] = v.z; ft[fr][fc * 4 + 3] = v.w;
    }
    __syncthreads();
    // ---- WMMA over the chunk: 16 steps of K=4 ----
    #pragma unroll
    for (int kk = 0; kk < BK; kk += 4) {
      v2f a, bb;
      a.x  = wt[r][kk + 2 * half];
      a.y  = wt[r][kk + 2 * half + 1];
      bb.x = ft[kk + 2 * half][n0 + r];
      bb.y = ft[kk + 2 * half + 1][n0 + r];
      acc = __builtin_amdgcn_wmma_f32_16x16x4_f32(
          false, a, false, bb, (short)0, acc, false, false);
    }
    __syncthreads();
  }

  // combined = (feats + neigh) * mask_row
  #pragma unroll
  for (int j = 0; j < 8; ++j) {
    int row = j + 8 * half;
    size_t o = ((size_t)(b * SVAL + m0 + row)) * HD + n0 + r;
    combined[o] = (feats[o] + acc[j]) * rMf[row];
  }
}

// ---------------------------------------------------------------------------
// out = f(x @ W + bias) over M x 128, K = 128 (W row-major 128x128).
//   LN:    layernorm (over 128) then gelu   (scalar-net first GEMM)
//   GELU:  gelu only                        (head hidden layers)
//   RESID: out = resid + x@W + bias         (scalar-net second GEMM)
// Block: 256 thr (8 waves) -> 16 rows x 128 cols tile.
// ---------------------------------------------------------------------------
template <bool LN, bool GELU, bool RESID>
__global__ __launch_bounds__(256) void k_gemm128(
    const float* __restrict__ x, const float* __restrict__ W,
    const float* __restrict__ bias,
    const float* __restrict__ ln_g, const float* __restrict__ ln_b,
    const float* __restrict__ resid, float* __restrict__ out) {
  __shared__ float xt[16][132];
  __shared__ float mu_s[16], rs_s[16];

  const int tid  = threadIdx.x;
  const int row0 = blockIdx.x * 16;

  const float4* x4 = (const float4*)(x + (size_t)row0 * HD);
  #pragma unroll
  for (int t = 0; t < 2; ++t) {
    int idx = tid + t * 256;
    int fr = idx >> 5, fc = idx & 31;
    float4 v = x4[fr * 32 + fc];
    xt[fr][fc * 4 + 0] = v.x; xt[fr][fc * 4 + 1] = v.y;
    xt[fr][fc * 4 + 2] = v.z; xt[fr][fc * 4 + 3] = v.w;
  }
  __syncthreads();

  const int lane = tid & 31;
  const int wv   = tid >> 5;
  const int half = lane >> 4;
  const int r    = lane & 15;
  const int n0   = wv * 16;

  v8f acc = {0.f,0.f,0.f,0.f,0.f,0.f,0.f,0.f};
  #pragma unroll
  for (int k = 0; k < HD; k += 4) {
    v2f a, bb;
    a.x  = xt[r][k + 2 * half];
    a.y  = xt[r][k + 2 * half + 1];
    bb.x = W[(k + 2 * half) * HD + n0 + r];
    bb.y = W[(k + 2 * half + 1) * HD + n0 + r];
    acc = __builtin_amdgcn_wmma_f32_16x16x4_f32(
        false, a, false, bb, (short)0, acc, false, false);
  }
  const float bcol = bias[n0 + r];

  if (LN) {
    __syncthreads();                       // done reading xt; reuse for h
    #pragma unroll
    for (int j = 0; j < 8; ++j) xt[j + 8 * half][n0 + r] = acc[j] + bcol;
    __syncthreads();
    if (tid < 16) {
      float sum = 0.f, sq = 0.f;
      for (int c = 0; c < HD; ++c) { float v = xt[tid][c]; sum += v; sq += v * v; }
      float mu  = sum * (1.0f / HD);
      float var = sq * (1.0f / HD) - mu * mu;
      mu_s[tid] = mu;
      rs_s[tid] = rsqrtf(var + 1e-6f);
    }
    __syncthreads();
    #pragma unroll
    for (int t = 0; t < 8; ++t) {
      int idx = tid + t * 256;
      int rr = idx >> 7, cc = idx & (HD - 1);
      float v = (xt[rr][cc] - mu_s[rr]) * rs_s[rr] * ln_g[cc] + ln_b[cc];
      out[(size_t)(row0 + rr) * HD + cc] = geluf(v);
    }
  } else {
    #pragma unroll
    for (int j = 0; j < 8; ++j) {
      int row = j + 8 * half;
      size_t o = (size_t)(row0 + row) * HD + n0 + r;
      float v = acc[j] + bcol;
      if (GELU)  v = geluf(v);
      if (RESID) v += resid[o];
      out[o] = v;
    }
  }
}

// ---------------------------------------------------------------------------
// Heads: scale = softplus(hidS @ W2s + b2s)+1e-6 ; trans = hidT @ W2t + b2t
// c2n = (c2*scale + trans)*m2 ; log_det += sum(log(scale))*m2
// ---------------------------------------------------------------------------
__global__ __launch_bounds__(256) void k_head(
    const float* __restrict__ hidS, const float* __restrict__ hidT,
    const float* __restrict__ W2s, const float* __restrict__ b2s,
    const float* __restrict__ W2t, const float* __restrict__ b2t,
    const float* __restrict__ coords, const unsigned char* __restrict__ mask,
    float* __restrict__ out) {
  __shared__ float red[256];
  const int atom = blockIdx.x * 256 + threadIdx.x;   // < B*s
  const int b = atom / SVAL;
  const int i = atom - b * SVAL;

  float s0 = b2s[0], s1 = b2s[1], s2 = b2s[2];
  float t0 = b2t[0], t1 = b2t[1], t2 = b2t[2];
  const float* hs = hidS + (size_t)atom * HD;
  const float* ht = hidT + (size_t)atom * HD;
  for (int k = 0; k < HD; ++k) {
    float a = hs[k], c = ht[k];
    s0 += a * W2s[k * 3 + 0]; s1 += a * W2s[k * 3 + 1]; s2 += a * W2s[k * 3 + 2];
    t0 += c * W2t[k * 3 + 0]; t1 += c * W2t[k * 3 + 1]; t2 += c * W2t[k * 3 + 2];
  }
  s0 = softplusf(s0) + 1e-6f;
  s1 = softplusf(s1) + 1e-6f;
  s2 = softplusf(s2) + 1e-6f;

  const int g2 = b * 2 * SVAL + SVAL + i;            // second-half atom index
  const float m2 = mask[g2] ? 1.0f : 0.0f;
  out[g2 * 3 + 0] = (coords[g2 * 3 + 0] * s0 + t0) * m2;
  out[g2 * 3 + 1] = (coords[g2 * 3 + 1] * s1 + t1) * m2;
  out[g2 * 3 + 2] = (coords[g2 * 3 + 2] * s2 + t2) * m2;

  red[threadIdx.x] = (logf(s0) + logf(s1) + logf(s2)) * m2;
  __syncthreads();
  for (int off = 128; off > 0; off >>= 1) {
    if (threadIdx.x < off) red[threadIdx.x] += red[threadIdx.x + off];
    __syncthreads();
  }
  if (threadIdx.x == 0)
    atomicAdd(out + (size_t)BVAL * 2 * SVAL * 3 + b, red[0]);
}

__global__ void k_copy_c1(const float* __restrict__ coords,
                          float* __restrict__ out) {
  int idx = blockIdx.x * blockDim.x + threadIdx.x;    // over B * s * 3
  const int per = SVAL * 3;
  if (idx >= BVAL * per) return;
  int b = idx / per, rr = idx - b * per;
  out[(size_t)b * 2 * per + rr] = coords[(size_t)b * 2 * per + rr];
}

__global__ void k_zero_logdet(float* __restrict__ out) {
  if (threadIdx.x < BVAL) out[(size_t)BVAL * 2 * SVAL * 3 + threadIdx.x] = 0.f;
}

// ---------------------------------------------------------------------------
extern "C" void kernel_launch(void* const* d_in, const int* in_sizes, int n_in,
                              void* d_out, int out_size, void* d_ws, size_t ws_size,
                              hipStream_t stream) {
  (void)in_sizes; (void)n_in; (void)out_size; (void)ws_size;

  const float*         coords = (const float*)d_in[0];
  const int*           types  = (const int*)d_in[1];
  const unsigned char* maskp  = (const unsigned char*)d_in[2]; // jax bool = 1B

  // params flattened in jax pytree order (dict keys sorted):
  // embed, layers[0..1]{W1,W2,b1,b2,ln_b,ln_g}, scale{W1,W2,b1,b2}, trans{...}
  const float* embed = (const float*)d_in[3];
  const float *W1[2], *W2[2], *b1[2], *b2[2], *lnb[2], *lng[2];
  for (int L = 0; L < 2; ++L) {
    int base = 4 + 6 * L;
    W1[L]  = (const float*)d_in[base + 0];
    W2[L]  = (const float*)d_in[base + 1];
    b1[L]  = (const float*)d_in[base + 2];
    b2[L]  = (const float*)d_in[base + 3];
    lnb[L] = (const float*)d_in[base + 4];
    lng[L] = (const float*)d_in[base + 5];
  }
  const float* W1s = (const float*)d_in[16];
  const float* W2s = (const float*)d_in[17];
  const float* b1s = (const float*)d_in[18];
  const float* b2s = (const float*)d_in[19];
  const float* W1t = (const float*)d_in[20];
  const float* W2t = (const float*)d_in[21];
  const float* b1t = (const float*)d_in[22];
  const float* b2t = (const float*)d_in[23];

  float* outp = (float*)d_out;
  float* feats = (float*)d_ws;                        // 8 MB
  float* bufA  = feats + (size_t)MROWS * HD;          // 8 MB
  float* bufB  = bufA  + (size_t)MROWS * HD;          // 8 MB

  const int total_feats = MROWS * HD;
  k_embed<<<(total_feats + 255) / 256, 256, 0, stream>>>(types, embed, feats,
                                                         total_feats);
  for (int L = 0; L < 2; ++L) {
    k_neigh<<<BVAL * (SVAL / 16), 256, 0, stream>>>(coords, maskp, feats, bufA);
    k_gemm128<true, true, false><<<MROWS / 16, 256, 0, stream>>>(
        bufA, W1[L], b1[L], lng[L], lnb[L], nullptr, bufB);
    k_gemm128<false, false, true><<<MROWS / 16, 256, 0, stream>>>(
        bufB, W2[L], b2[L], nullptr, nullptr, feats, feats);
  }
  k_gemm128<false, true, false><<<MROWS / 16, 256, 0, stream>>>(
      feats, W1s, b1s, nullptr, nullptr, nullptr, bufA);
  k_gemm128<false, true, false><<<MROWS / 16, 256, 0, stream>>>(
      feats, W1t, b1t, nullptr, nullptr, nullptr, bufB);

  k_zero_logdet<<<1, 32, 0, stream>>>(outp);
  k_head<<<MROWS / 256, 256, 0, stream>>>(bufA, bufB, W2s, b2s, W2t, b2t,
                                          coords, maskp, outp);
  k_copy_c1<<<(MROWS * 3 + 255) / 256, 256, 0, stream>>>(coords, outp);
}